// MeanDistanceFromRecoToTrue_20298015441365
// MI455X (gfx1250) — compile-verified
//
#include <hip/hip_runtime.h>

// ---------------------------------------------------------------------------
// MeanDistanceFromRecoToTrue on gfx1250 (MI455X, wave32)
//
//   d2(p,t) = ||p||^2 + ||t||^2 - 2 p.t  via V_WMMA_F32_16X16X4_F32:
//     A row m  = [-2*pz, -2*py, -2*px, 1.0]          (16x4)
//     B col n  = [  tz ,   ty ,   tx , ||t||^2 ]     (4x16)
//     C = 0 (inline constant)  =>  D[m,n] = ||t_n||^2 - 2 p_m . t_n
//   min over n, then + ||p_m||^2, sqrt, masked sum.  All operands are small
//   integers so every fp32 op inside the WMMA is exact (and never NaN).
// ---------------------------------------------------------------------------

#define VOX   64000      // 40*40*40
#define GW    40u
#define GHW   1600u
#define KP    2048
#define KT    40960
#define KPP   (KP + 16)  // padded capacities (room for 16 sentinel records)
#define KTP   (KT + 16)
#define NTILE 128        // KP / 16
#define EPS_T 2.0f
#define BIG   1e30f

typedef __attribute__((ext_vector_type(2))) float v2f;
typedef __attribute__((ext_vector_type(8))) float v8f;

// Raw v_min_num_f32 (skip fminf's NaN-canonicalize: values are never NaN here)
__device__ __forceinline__ float fmin_raw(float a, float b) {
    float r;
    asm("v_min_num_f32 %0, %1, %2" : "=v"(r) : "v"(a), "v"(b));
    return r;
}

// ---------------------------------------------------------------------------
// Kernel 1: in-order stream compaction (matches jnp.nonzero(size=K): first K
// passing indices in ascending order). One block per batch. Emits per-point
// records float4{z, y, x, norm2} and appends 16 sentinel records.
// cnts layout: [p_cnt_b0, p_cnt_b1, t_cnt_b0, t_cnt_b1]
// ---------------------------------------------------------------------------
__device__ __forceinline__ float4 voxel_record(unsigned u) {
    const unsigned z = u / GHW;
    const unsigned r = u % GHW;
    const unsigned y = r / GW;
    const unsigned x = r % GW;
    const float fz = (float)z, fy = (float)y, fx = (float)x;
    return make_float4(fz, fy, fx, fz * fz + fy * fy + fx * fx);
}

__global__ __launch_bounds__(1024)
void compact_kernel(const float* __restrict__ inp, const float* __restrict__ tgt,
                    float4* __restrict__ p_pts, float4* __restrict__ t_pts,
                    int* __restrict__ cnts) {
    const int b = blockIdx.x;
    const float* x = inp + b * VOX;
    const float* t = tgt + b * VOX;
    float4* po = p_pts + (size_t)b * KPP;
    float4* to = t_pts + (size_t)b * KTP;

    __shared__ unsigned swp[32], swt[32];     // per-wave totals
    __shared__ unsigned sofp[32], soft[32];   // per-wave exclusive offsets
    __shared__ unsigned baseP, baseT;         // running totals (in-order)

    const int lane = threadIdx.x & 31;
    const int wid  = threadIdx.x >> 5;
    if (threadIdx.x == 0) { baseP = 0u; baseT = 0u; }
    __syncthreads();

    for (int start = 0; start < VOX; start += 1024) {
        const int i = start + (int)threadIdx.x;
        const bool pp = (i < VOX) && (x[i] > EPS_T);
        const bool tp = (i < VOX) && (t[i] > 0.0f);
        const unsigned mp = (unsigned)__ballot(pp);
        const unsigned mt = (unsigned)__ballot(tp);
        if (lane == 0) { swp[wid] = (unsigned)__popc(mp); swt[wid] = (unsigned)__popc(mt); }
        __syncthreads();
        if (threadIdx.x == 0) {
            unsigned aP = baseP, aT = baseT;
            for (int w = 0; w < 32; ++w) {
                sofp[w] = aP; aP += swp[w];
                soft[w] = aT; aT += swt[w];
            }
            baseP = aP; baseT = aT;
        }
        __syncthreads();
        const unsigned lm = (1u << lane) - 1u;
        const unsigned posP = sofp[wid] + (unsigned)__popc(mp & lm);
        const unsigned posT = soft[wid] + (unsigned)__popc(mt & lm);
        if (pp && posP < (unsigned)KP) po[posP] = voxel_record((unsigned)i);
        if (tp && posT < (unsigned)KT) to[posT] = voxel_record((unsigned)i);
        __syncthreads();
    }
    // sentinel padding so the NN loop never needs bounds checks
    if (threadIdx.x < 16) {
        const unsigned pc = baseP < (unsigned)KP ? baseP : (unsigned)KP;
        const unsigned tc = baseT < (unsigned)KT ? baseT : (unsigned)KT;
        po[pc + threadIdx.x] = make_float4(0.0f, 0.0f, 0.0f, 0.0f);
        to[tc + threadIdx.x] = make_float4(0.0f, 0.0f, 0.0f, BIG);
        if (threadIdx.x == 0) {
            cnts[b]     = (int)pc;
            cnts[2 + b] = (int)tc;
        }
    }
}

// ---------------------------------------------------------------------------
// Kernel 2: per pred-tile 1-NN via WMMA. Block = 8 waves; each wave strides
// over 16-wide true tiles. Record layout lets each lane load exactly its
// operand half: half 0 -> float2{z,y} (K=0,1), half 1 -> float2{x,n2} (K=2,3).
// ---------------------------------------------------------------------------
__global__ __launch_bounds__(256)
void nn_kernel(const float4* __restrict__ p_pts, const float4* __restrict__ t_pts,
               const int* __restrict__ cnts, float* __restrict__ tile_sums) {
    const int tile = blockIdx.x;          // 0..NTILE-1
    const int b    = blockIdx.y;          // 0..1
    const int pcnt = cnts[b];
    const int tcnt = cnts[2 + b];
    float* out = tile_sums + b * NTILE + tile;

    const int tid  = (int)threadIdx.x;
    const int lane = tid & 31;
    // wave id is wave-uniform: hoist to an SGPR so loop control is scalar
    const int wid  = __builtin_amdgcn_readfirstlane(tid >> 5);
    const int base = tile * 16;

    __shared__ float smin[8][16];
    __shared__ float ssum[16];

    if (base >= pcnt || tcnt == 0) {      // block-uniform condition
        if (tid == 0) *out = 0.0f;
        return;
    }

    const int m    = lane & 15;           // M (and N) index within tile
    const int half = lane >> 4;           // 0: K=0,1   1: K=2,3

    // ---- A tile (16x4): lanes 0-15 hold K=0,1; lanes 16-31 hold K=2,3.
    // Record for pred point base+m is always defined (sentinel padded).
    const float2* pa = (const float2*)(p_pts + (size_t)b * KPP + base);
    const float2 ea = pa[2 * m + half];   // {z,y} or {x,n2}
    v2f A;
    A.x = -2.0f * ea.x;                          // K=0: -2z   | K=2: -2x
    A.y = half ? 1.0f : (-2.0f * ea.y);          // K=1: -2y   | K=3:  1.0

    const v8f Cz = {};                    // folds to WMMA inline-constant 0

    float minv[8];
#pragma unroll
    for (int r = 0; r < 8; ++r) minv[r] = BIG;

    const int ttiles = (tcnt + 15) >> 4;
    const float2* tq = (const float2*)(t_pts + (size_t)b * KTP);

#pragma unroll 4
    for (int k = wid; k < ttiles; k += 8) {          // wave-uniform loop
        // B tile (4x16): lane -> col n = lane&15; half selects K rows.
        const float2 e = tq[(k * 16 + m) * 2 + half]; // {z,y} | {x,n2}
        v2f B;
        B.x = e.x;
        B.y = e.y;
        // D[m_r, n] = ||t_n||^2 - 2 * p_{m_r} . t_n   (exact in fp32)
        v8f D = __builtin_amdgcn_wmma_f32_16x16x4_f32(
            /*neg_a=*/false, A, /*neg_b=*/false, B,
            /*c_mod=*/(short)0, Cz, /*reuse_a=*/false, /*reuse_b=*/false);
#pragma unroll
        for (int r = 0; r < 8; ++r) minv[r] = fmin_raw(minv[r], D[r]);
    }

    // min over n within each 16-lane half (C/D layout: lane -> N = lane&15)
#pragma unroll
    for (int off = 1; off < 16; off <<= 1) {
#pragma unroll
        for (int r = 0; r < 8; ++r)
            minv[r] = fmin_raw(minv[r], __shfl_xor(minv[r], off, 32));
    }
    // C/D layout: VGPR r holds M=r (lanes 0-15) and M=r+8 (lanes 16-31)
    if ((lane & 15) == 0) {
#pragma unroll
        for (int r = 0; r < 8; ++r) smin[wid][half * 8 + r] = minv[r];
    }
    __syncthreads();

    if (tid < 16) {
        float mn = BIG;
#pragma unroll
        for (int w = 0; w < 8; ++w) mn = fmin_raw(mn, smin[w][tid]);
        float s = 0.0f;
        if (base + tid < pcnt) {
            const float pn2 = pa[2 * tid + 1].y;        // ||p||^2 from record
            s = sqrtf(fmaxf(mn + pn2, 0.0f));           // exact d2 >= 0
        }
        ssum[tid] = s;
    }
    __syncthreads();
    if (tid == 0) {
        float acc = 0.0f;
        for (int i = 0; i < 16; ++i) acc += ssum[i];    // fixed order
        *out = acc;
    }
}

// ---------------------------------------------------------------------------
// Kernel 3: deterministic final reduction + mean.
// ---------------------------------------------------------------------------
__global__ void finalize_kernel(const float* __restrict__ tile_sums,
                                const int* __restrict__ cnts,
                                float* __restrict__ out) {
    if (threadIdx.x == 0 && blockIdx.x == 0) {
        float total = 0.0f;
        for (int i = 0; i < 2 * NTILE; ++i) total += tile_sums[i];
        const int cnt = cnts[0] + cnts[1];
        out[0] = (cnt > 0) ? (total / (float)cnt) : 0.0f;
    }
}

// ---------------------------------------------------------------------------
extern "C" void kernel_launch(void* const* d_in, const int* in_sizes, int n_in,
                              void* d_out, int out_size, void* d_ws, size_t ws_size,
                              hipStream_t stream) {
    (void)in_sizes; (void)n_in; (void)out_size; (void)ws_size;
    const float* inp = (const float*)d_in[0];
    const float* tgt = (const float*)d_in[1];

    char* ws = (char*)d_ws;
    float4* p_pts = (float4*)(ws);                                  // 2*KPP float4
    float4* t_pts = (float4*)(ws + (size_t)2 * KPP * 16);           // 2*KTP float4
    char*   tail  = ws + (size_t)2 * KPP * 16 + (size_t)2 * KTP * 16;
    int*    cnts      = (int*)tail;                                 // 4 ints
    float*  tile_sums = (float*)(tail + 16);                        // 2*NTILE floats

    compact_kernel<<<2, 1024, 0, stream>>>(inp, tgt, p_pts, t_pts, cnts);

    dim3 grid(NTILE, 2);
    nn_kernel<<<grid, 256, 0, stream>>>(p_pts, t_pts, cnts, tile_sums);

    finalize_kernel<<<1, 32, 0, stream>>>(tile_sums, cnts, (float*)d_out);
}